// MAPKPI3KODE_68393059221800
// MI455X (gfx1250) — compile-verified
//
#include <hip/hip_runtime.h>
#include <cstdint>

// MAPK/PI3K ODE RHS for MI455X (gfx1250).
// Memory-bound: 285 MB traffic -> ~12 us floor @ 23.3 TB/s; ~0.4 GFLOP fp32 is negligible.
// Strategy: coalesced async VMEM->LDS tile staging (CDNA5 ASYNCcnt path),
// LDS row reads, interleaved compute + non-temporal b128 stores,
// uniform transcendentals hoisted to a 1-thread prep kernel.

#define EPSF 1e-10f
#define NSTATE 68
#define TPB 128  // systems per block; tile = 128*68*4 = 34,816 B LDS (< 320 KB WGP)

typedef float v4f __attribute__((ext_vector_type(4)));

// ---------------- prep: copy params + derived uniform constants into d_ws ----
__global__ void mapk_prep(const float* __restrict__ params, float* __restrict__ C) {
  if (threadIdx.x == 0 && blockIdx.x == 0) {
    float p[75];
    for (int i = 0; i < 75; ++i) { p[i] = params[i]; C[i] = p[i]; }
    float IC50n = powf(p[61], p[62]);      // IC50_vem ** Hill_n_vem
    float Vemn  = powf(p[57], p[62]);      // Vemurafenib ** Hill_n_vem
    C[75] = p[0] * IC50n / (IC50n + Vemn + EPSF);   // kBRAF_eff
    float Ktn = powf(p[55], p[56]);        // K_tram_KSR ** n_tram
    float tn  = powf(p[53], p[56]);        // Tram ** n_tram
    C[76] = Ktn / (Ktn + tn + EPSF);                // tram_ksr
    C[77] = p[27] / (p[22] + EPSF);                 // km_Dusp / kDusps
    C[78] = p[28] / (p[23] + EPSF);                 // km_Sprty / kSproutyForm
  }
}

// ---------------- main RHS kernel -------------------------------------------
__global__ __launch_bounds__(TPB) void mapk_rhs(const float* __restrict__ yg,
                                                const float* __restrict__ C,
                                                float* __restrict__ dout,
                                                int B) {
  __shared__ float sy[TPB * NSTATE];

  const int tid = threadIdx.x;
  const size_t blockBase = (size_t)blockIdx.x * (TPB * NSTATE);  // floats
  const size_t total = (size_t)B * NSTATE;

  // --- Stage tile into LDS with coalesced async vmem->LDS b128 copies ---
#pragma unroll
  for (int k = 0; k < NSTATE / 4; ++k) {            // 17 float4s per thread
    const int idx4 = tid + k * TPB;                 // linear float4 index in tile
    const size_t goff = blockBase + (size_t)idx4 * 4;
    if (goff < total) {
      const uint64_t gaddr = (uint64_t)(uintptr_t)(yg + goff);
      const uint32_t laddr = (uint32_t)(uintptr_t)(&sy[idx4 * 4]);  // LDS byte offset
      asm volatile("global_load_async_to_lds_b128 %0, %1, off"
                   :: "v"(laddr), "v"(gaddr) : "memory");
    }
  }
  asm volatile("s_wait_asynccnt 0" ::: "memory");
  __syncthreads();

  const int sys = blockIdx.x * TPB + tid;
  if (sys >= B) return;

  // --- Read my row from LDS (ds_load_b128) and clip to >= 0 ---
  float Y[NSTATE];
  {
    const v4f* sp = (const v4f*)(&sy[tid * NSTATE]);
#pragma unroll
    for (int k = 0; k < NSTATE / 4; ++k) {
      v4f q = sp[k];
      Y[4 * k + 0] = fmaxf(q.x, 0.f);
      Y[4 * k + 1] = fmaxf(q.y, 0.f);
      Y[4 * k + 2] = fmaxf(q.z, 0.f);
      Y[4 * k + 3] = fmaxf(q.w, 0.f);
    }
  }

  // --- Uniform parameters (scalar loads) ---
  const float ka1=C[0], kr1=C[1], kc1=C[2], kpCraf=C[3], kpMek=C[4], kpErk=C[5];
  const float kDegradEgfr=C[6], kErkInbEgfr=C[7], kShcDephos=C[8], kptpDeg=C[9];
  const float kGrb2CombShc=C[10], kSprtyInbGrb2=C[11], kSosCombGrb2=C[12];
  const float kErkPhosSos=C[13], kErkPhosPcraf=C[14], kPcrafDegrad=C[15];
  const float kErkPhosMek=C[16], kMekDegrad=C[17], kDuspInbErk=C[18], kErkDeg=C[19];
  const float kinbBraf=C[20], kDuspStop=C[21], kSprtyComeDown=C[24], kdegrad=C[25];
  const float km_Dusp=C[27], km_Sprty=C[28], kErkDephos=C[29], kDuspDeg=C[30];
  const float kHer2=C[31], kHer3=C[32], kbEGFR=C[33], kbHer2=C[34], kbHer3=C[35];
  const float kbIGFR=C[36], kUnbind=C[37], kPI3K=C[38], kMTORfb=C[39], kPIP=C[40];
  const float kPTEN=C[41], kAkt=C[42], kdegAKT=C[43], kb1=C[44], k43b1=C[45];
  const float k4ebp1=C[46], k4e_de=C[47], kKSRp=C[48], kKSRd=C[49];
  const float kMekBr=C[50], kMekCr=C[51], kMekKSR=C[52];
  const float Vem=C[57], kDimF=C[58], kDimD=C[59], kParadox=C[60];
  const float kPDGFR=C[63], kbPDGFR=C[64], kS6Kp=C[65], kS6Kd=C[66];
  const float kRASPI3K=C[67], kERKIRS=C[68], kERKPTEN=C[69], kAKTCRAF=C[70];
  const float kS6KIRS=C[71], kERKGAB1=C[72], kAKTTSC2=C[73], kERKRSK=C[74];
  const float kBRAF_eff=C[75], tram_ksr=C[76], duspRatio=C[77], spryRatio=C[78];

  float* gout = dout + (size_t)sys * NSTATE;
#define STORE4(i, a, b, c, d)                                    \
  { v4f t_ = {(a), (b), (c), (d)};                               \
    __builtin_nontemporal_store(t_, (v4f*)(gout + (i))); }

  const float y28 = Y[28];

  // --- EGFR / Her2 / Her3 ---
  float d0 = -ka1 * Y[0] + kr1 * Y[1];
  float d1 =  ka1 * Y[0] - kr1 * Y[1] - kc1 * Y[1];
  float d2 =  kc1 * Y[1] - kDegradEgfr * Y[2] - kErkInbEgfr * y28 * Y[2];
  float d3 = -kHer2 * Y[3] + kr1 * Y[4];
  STORE4(0, d0, d1, d2, d3);
  float d4 =  kHer2 * Y[3] - kr1 * Y[4] - kc1 * Y[4];
  float d5 =  kc1 * Y[4] - kDegradEgfr * Y[5] - kErkInbEgfr * y28 * Y[5];
  float d6 = -kHer3 * Y[6] + kr1 * Y[7];
  float d7 =  kHer3 * Y[6] - kr1 * Y[7] - kc1 * Y[7];
  STORE4(4, d4, d5, d6, d7);
  // --- Shc / Grb2 / Sos / Ras ---
  float d8  =  kc1 * Y[7] - kDegradEgfr * Y[8] - kErkInbEgfr * y28 * Y[8];
  float d9  = -ka1 * Y[2] * Y[9];
  float d10 =  ka1 * Y[2] * Y[9] - kShcDephos * Y[11] * Y[10];
  float d11 = -kptpDeg * Y[10] * Y[11];
  STORE4(8, d8, d9, d10, d11);
  float d12 = kGrb2CombShc * Y[10] * Y[2] - kSprtyInbGrb2 * Y[26] * Y[12];
  float d13 = kSosCombGrb2 * Y[12] * Y[10] - kErkPhosSos * Y[24] * Y[13];
  float d14 = -ka1 * Y[13] * Y[14];
  float d15 =  ka1 * Y[13] * Y[14];
  STORE4(12, d12, d13, d14, d15);
  float d16 = -ka1 * Y[13] * Y[16];
  float d17 =  ka1 * Y[13] * Y[16];
  float d18 = -ka1 * Y[13] * Y[18];
  float d19 =  ka1 * Y[13] * Y[18] - ka1 * Y[19] * Y[20];
  STORE4(16, d16, d17, d18, d19);
  float d20 = -ka1 * Y[19] * Y[20];
  // --- RAF with vemurafenib paradox ---
  float dimF = kDimF * Y[24] * Y[21] * Vem;
  float dimD = kDimD * Y[61];
  float d21 = -kpCraf * Y[19] * Y[21] + kErkPhosPcraf * y28 * Y[22]
              + kPcrafDegrad * Y[22] * Y[35] - dimF + dimD
              - kAKTCRAF * Y[52] * Y[21];
  float d22 =  kpCraf * Y[19] * Y[21] - kErkPhosPcraf * y28 * Y[22]
              - kPcrafDegrad * Y[22] * Y[35] + kParadox * Vem * Y[61];
  float d23 = -kBRAF_eff * Y[23] * Y[19] - dimF + dimD;
  STORE4(20, d20, d21, d22, d23);
  float d24 =  kBRAF_eff * Y[23] * Y[19] - kinbBraf * Y[24] - dimF + dimD;
  float d61 =  dimF - dimD - kPcrafDegrad * Y[61] * Y[35];   // stored in group 60..63
  // --- MEK / ERK ---
  float raf_to_mek = kpMek * Y[22] + kMekBr * Y[24] + kMekCr * Y[22];
  float mek_act = (raf_to_mek + kMekKSR * Y[60]) * Y[25];
  float d25 = -mek_act + kErkPhosMek * y28 * Y[26] + kMekDegrad * Y[26] * Y[34];
  float d26 =  mek_act - kErkPhosMek * y28 * Y[26] - kMekDegrad * Y[26] * Y[34];
  float d27 = -kpErk * Y[26] * Y[27] + kDuspInbErk * Y[30] * y28
              + kErkDeg * y28 * Y[33] + kErkDephos * Y[30] * y28;
  STORE4(24, d24, d25, d26, d27);
  float d28 =  kpErk * Y[26] * Y[27] - kDuspInbErk * Y[30] * y28
              - kErkDeg * y28 * Y[33] - kErkDephos * Y[30] * y28;
  // --- DUSP / Sprouty feedback ---
  float d29 = km_Dusp * y28 / ((1.f + duspRatio * y28) + EPSF)
              - kDuspStop * Y[29] * Y[36] - kDuspDeg * Y[29] * y28;
  float d30 = -kDuspStop * Y[29] * Y[30];
  float d31 = km_Sprty * y28 / ((1.f + spryRatio * y28) + EPSF)
              - kSprtyComeDown * Y[31] * Y[32];
  STORE4(28, d28, d29, d30, d31);
  float d32 = -kSprtyComeDown * Y[31] * Y[32];
  float d33 = -kErkDeg * y28 * Y[33];
  float d34 = -kMekDegrad * Y[26] * Y[34];
  float d35 = -kPcrafDegrad * Y[22] * Y[35];
  STORE4(32, d32, d33, d34, d35);
  float d36 = -kDuspStop * Y[29] * Y[36];
  // --- IGFR / IRS ---
  float d37 = -ka1 * Y[37] + kr1 * Y[38];
  float d38 =  ka1 * Y[37] - kr1 * Y[38] - kc1 * Y[38];
  float d39 =  kc1 * Y[38] - kErkInbEgfr * y28 * Y[39];
  STORE4(36, d36, d37, d38, d39);
  float erk_irs = kERKIRS * y28 * Y[41];
  float s6k_irs = kS6KIRS * Y[66] * Y[41];
  float d40 = -ka1 * Y[2] * Y[40] + erk_irs + s6k_irs;
  float d41 =  ka1 * Y[2] * Y[40] - erk_irs - s6k_irs;
  // --- p85 binding ---
  float gab1 = 1.f / (1.f + kERKGAB1 * y28);
  float bE  = kbEGFR * Y[2] * Y[42] * gab1;
  float bH2 = kbHer2 * Y[5] * Y[42] * gab1;
  float bH3 = kbHer3 * Y[8] * Y[42] * gab1;
  float bI  = kbIGFR * Y[39] * Y[42];
  float bP  = kbPDGFR * Y[64] * Y[42];
  float tot85 = Y[43] + Y[44] + Y[45] + Y[46] + Y[67];
  float d42 = -bE - bH2 - bH3 - bI - bP + kUnbind * tot85;
  float d43 = bE - kUnbind * Y[43];
  STORE4(40, d40, d41, d42, d43);
  float d44 = bH2 - kUnbind * Y[44];
  float d45 = bH3 - kUnbind * Y[45];
  float d46 = bI - kUnbind * Y[46];
  float d67 = bP - kUnbind * Y[67];                 // stored in group 64..67
  // --- PI3K / AKT / mTOR ---
  float pi3k = kPI3K * tot85 * Y[47] + kRASPI3K * Y[15] * Y[47];
  float mtfb = kMTORfb * Y[56] * Y[48];
  float d47 = -pi3k + mtfb;
  STORE4(44, d44, d45, d46, d47);
  float d48 =  pi3k - mtfb;
  float pip  = kPIP * Y[48] * Y[49];
  float pten = kPTEN * Y[51] * Y[50];
  float d49 = -pip + pten;
  float d50 =  pip - pten;
  float d51 = kERKPTEN * y28 - kdegrad * Y[51];
  STORE4(48, d48, d49, d50, d51);
  float akt = kAkt * Y[50] * Y[53];
  float d52 =  akt - kdegAKT * Y[52];
  float d53 = -akt + kdegAKT * Y[52];
  float tsc = kAKTTSC2 * Y[52] * Y[54];
  float d54 = -tsc;
  float d55 =  tsc - kdegrad * Y[55];
  STORE4(52, d52, d53, d54, d55);
  float fb1 = kb1 * Y[52] * Y[57];
  float d56 =  fb1 - k43b1 * Y[56];
  float d57 = -fb1 + k43b1 * Y[56];
  float e4 = k4ebp1 * Y[56] * Y[58];
  float d58 = -e4 + k4e_de * Y[59];
  float d59 =  e4 - k4e_de * Y[59];
  STORE4(56, d56, d57, d58, d59);
  // --- KSR (trametinib), PDGFR, S6K ---
  float ksrp = kKSRp * Y[19] * Y[62] * tram_ksr;
  float d60 =  ksrp - kKSRd * Y[60];
  float d62 = -ksrp + kKSRd * Y[60];
  float d63 = -kPDGFR * Y[63];
  STORE4(60, d60, d61, d62, d63);
  float d64 = kPDGFR * Y[63] - kDegradEgfr * Y[64];
  float s6  = kS6Kp * Y[56] * Y[65];
  float rsk = kERKRSK * y28 * Y[65];
  float d65 = -s6 + kS6Kd * Y[66] - rsk;
  float d66 =  s6 - kS6Kd * Y[66] + rsk;
  STORE4(64, d64, d65, d66, d67);
#undef STORE4
}

extern "C" void kernel_launch(void* const* d_in, const int* in_sizes, int n_in,
                              void* d_out, int out_size, void* d_ws, size_t ws_size,
                              hipStream_t stream) {
  (void)n_in; (void)out_size; (void)ws_size;
  // setup_inputs order: t (1), y (B*68), params (75)
  const float* y      = (const float*)d_in[1];
  const float* params = (const float*)d_in[2];
  float* out = (float*)d_out;
  float* C   = (float*)d_ws;  // 79 floats of uniform constants
  const int B = in_sizes[1] / NSTATE;

  mapk_prep<<<1, 32, 0, stream>>>(params, C);
  const int blocks = (B + TPB - 1) / TPB;
  mapk_rhs<<<blocks, TPB, 0, stream>>>(y, C, out, B);
}